// SimtaPlusLayer_28793460752552
// MI455X (gfx1250) — compile-verified
//
#include <hip/hip_runtime.h>

typedef float v2f __attribute__((ext_vector_type(2)));
typedef float v8f __attribute__((ext_vector_type(8)));

#define B_       32
#define T_       1024
#define TEMP_IN  128
#define STAT_IN  64
#define CIN      192          // TEMP_IN + STAT_IN
#define C_       128          // TEMP_OUT
#define STATOUT  64
#define EPS      1e-5f
#define NEGSLOPE 0.01f
#define SCALE    0.08838834764831845f              // 1/sqrt(128)
#define SCALE_X  (SCALE * 1.44269504088896340736f) // fold log2(e) into the scale

#define BM 128                // rows per block
#define BN 64                 // columns per tile
#define NTILE (T_ / BN)       // 16
#define KT_STRIDE 132         // P column tile [BN][128] padded (132%64==4 -> bank spread)
#define PT_STRIDE 68          // probs/ta tiles [BM][64] padded (68%64==4)
// dynamic LDS layout (floats): Kt0 | Kt1 | Ta0 | Ta1 | Pe
#define KT_F   (BN * KT_STRIDE)   // 8448
#define PT_F   (BM * PT_STRIDE)   // 8704
#define SMEM_F (2 * KT_F + 3 * PT_F)  // 43008 floats = 168 KB

// ---------------------------------------------------------------- zero fill
__global__ void zero_kernel(float* __restrict__ p, int n) {
    int i = blockIdx.x * blockDim.x + threadIdx.x;
    if (i < n) p[i] = 0.0f;
}

// ------------------------------------------- linear1 + BN + leaky + scatter
__global__ __launch_bounds__(128)
void fused_linear_scatter(const float* __restrict__ x_temp,
                          const float* __restrict__ x_stat,
                          const int*   __restrict__ batch_idx,
                          const int*   __restrict__ pos_idx,
                          const float* __restrict__ W1,
                          const float* __restrict__ b1,
                          const float* __restrict__ gamma1,
                          const float* __restrict__ beta1,
                          const float* __restrict__ mean1,
                          const float* __restrict__ var1,
                          float* __restrict__ padded, int n_rows) {
    __shared__ float xs[16][CIN];                       // 12 KB
    const int row0 = blockIdx.x * 16;
    const int tid  = threadIdx.x;                       // 0..127 (out channel)

    for (int r = 0; r < 16; ++r) {
        int n = row0 + r;
        for (int c = tid; c < CIN; c += 128) {
            float v = 0.0f;
            if (n < n_rows) {
                v = (c < TEMP_IN)
                        ? x_temp[(size_t)n * TEMP_IN + c]
                        : x_stat[(size_t)batch_idx[n] * STAT_IN + (c - TEMP_IN)];
            }
            xs[r][c] = v;
        }
    }
    __syncthreads();

    float acc[16];
#pragma unroll
    for (int r = 0; r < 16; ++r) acc[r] = 0.0f;

    const float* wrow = W1 + (size_t)tid * CIN;
    for (int k = 0; k < CIN; ++k) {
        float w = wrow[k];
#pragma unroll
        for (int r = 0; r < 16; ++r) acc[r] = fmaf(xs[r][k], w, acc[r]);
    }

    const float sc = gamma1[tid] * rsqrtf(var1[tid] + EPS);
    const float bb = b1[tid];
#pragma unroll
    for (int r = 0; r < 16; ++r) {
        int n = row0 + r;
        if (n < n_rows) {
            float h = (acc[r] + bb - mean1[tid]) * sc + beta1[tid];
            h = (h >= 0.0f) ? h : NEGSLOPE * h;
            padded[((size_t)batch_idx[n] * T_ + pos_idx[n]) * C_ + tid] = h;
        }
    }
}

// ------------------------------------------------------------- stat branch
__global__ __launch_bounds__(64)
void stat_kernel(const float* __restrict__ x_stat,
                 const float* __restrict__ W2, const float* __restrict__ b2,
                 const float* __restrict__ gamma2, const float* __restrict__ beta2,
                 const float* __restrict__ mean2, const float* __restrict__ var2,
                 float* __restrict__ out) {
    const int b = blockIdx.x;
    const int o = threadIdx.x;                          // 0..63
    float acc = 0.0f;
    const float* xr = x_stat + (size_t)b * STAT_IN;
    const float* wr = W2 + (size_t)o * STAT_IN;
    for (int k = 0; k < STAT_IN; ++k) acc = fmaf(xr[k], wr[k], acc);
    float h = (acc + b2[o] - mean2[o]) * (gamma2[o] * rsqrtf(var2[o] + EPS)) + beta2[o];
    h = (h >= 0.0f) ? h : NEGSLOPE * h;
    out[(size_t)b * STATOUT + o] = h;
}

// ----------------------- async (ASYNCcnt) global->LDS tile prefetch issue
__device__ __forceinline__ void issue_tile_loads(uint32_t smem_base, int buf,
                                                 const float* __restrict__ Pb,
                                                 const float* __restrict__ TAb,
                                                 int t0, int s0, int tid) {
    // P column tile: BN x 128 floats -> 2048 x 16B chunks, 8 per thread
    uint32_t kbase = smem_base + (uint32_t)(buf * (KT_F * 4));
#pragma unroll
    for (int i = 0; i < 8; ++i) {
        int c = tid + i * 256;            // 0..2047
        int r = c >> 5, q = c & 31;       // 32 chunks per 512B row
        asm volatile("global_load_async_to_lds_b128 %0, %1, off"
                     :: "v"(kbase + (uint32_t)(r * (KT_STRIDE * 4) + q * 16)),
                        "v"(Pb + (size_t)(s0 + r) * C_ + q * 4)
                     : "memory");
    }
    // temp_attn tile: BM x 64 floats -> 2048 x 16B chunks; stream non-temporal
    uint32_t tbase = smem_base + (uint32_t)(2 * KT_F * 4) + (uint32_t)(buf * (PT_F * 4));
#pragma unroll
    for (int i = 0; i < 8; ++i) {
        int c = tid + i * 256;
        int r = c >> 4, q = c & 15;       // 16 chunks per 256B row
        asm volatile("global_load_async_to_lds_b128 %0, %1, off th:TH_LOAD_NT"
                     :: "v"(tbase + (uint32_t)(r * (PT_STRIDE * 4) + q * 16)),
                        "v"(TAb + (size_t)(t0 + r) * T_ + s0 + q * 4)
                     : "memory");
    }
}

// --------------------------------------------------- fused dense attention
// One block: batch b, 128 rows, 8 waves (16-row strip each). Single pass over
// 16 column tiles with async double-buffered LDS staging:
//   S = Q*Kt^T (WMMA f32), e = exp2(S*scale*log2e), Z += rowsum(e),
//   Oe += e @ P_tile, Ot += ta @ P_tile   (shared B frags)
// Final: out = Oe/Z + Ot, written straight into packed layout.
__global__ __launch_bounds__(256)
void attn_kernel(const float* __restrict__ padded,      // [B,T,C]
                 const float* __restrict__ temp_attn,   // [B,T,T]
                 const long long* __restrict__ lengths, // [B]
                 float* __restrict__ x_out) {           // [N,C] packed
    extern __shared__ float smem[];
    const uint32_t smem_base = (uint32_t)(uintptr_t)smem;   // LDS byte offset
    float* PeP = smem + 2 * KT_F + 2 * PT_F;                // probs staging

    const int b    = blockIdx.x >> 3;
    const int t0   = (blockIdx.x & 7) * BM;
    const int tid  = threadIdx.x;
    const int lane = tid & 31;
    const int wv   = tid >> 5;          // wave 0..7
    const int l16  = lane & 15;
    const int lh   = lane >> 4;         // half: 0/1

    const float* Pb  = padded    + (size_t)b * T_ * C_;
    const float* TAb = temp_attn + (size_t)b * T_ * T_;
    const float* Qrow = Pb + (size_t)(t0 + wv * 16 + l16) * C_;  // A-layout row

    const v8f z8 = {0.f, 0.f, 0.f, 0.f, 0.f, 0.f, 0.f, 0.f};
    v8f Oe[8], Ot[8];
#pragma unroll
    for (int j = 0; j < 8; ++j) { Oe[j] = z8; Ot[j] = z8; }
    float zacc[8] = {0.f, 0.f, 0.f, 0.f, 0.f, 0.f, 0.f, 0.f};

    // prologue: prefetch tile 0 into buffer 0
    issue_tile_loads(smem_base, 0, Pb, TAb, t0, 0, tid);

    for (int it = 0; it < NTILE; ++it) {
        const int cur = it & 1;
        // current tile's async loads complete, then make visible to all waves
        asm volatile("s_wait_asynccnt 0x0" ::: "memory");
        __syncthreads();
        // prefetch next tile into the other buffer (last read in the previous iteration)
        if (it + 1 < NTILE)
            issue_tile_loads(smem_base, 1 - cur, Pb, TAb, t0, (it + 1) * BN, tid);

        const float* Ktc = smem + cur * KT_F;
        const float* Tac = smem + 2 * KT_F + cur * PT_F;

        // ---- S = Q Kt^T (16 x BN per wave), K-dim = 128 in chunks of 4
        //      A frags stream from L1-hot global (keeps VGPR pressure low),
        //      B frags from padded LDS (conflict-free ds_load_2addr_b64)
        v8f S[4];
#pragma unroll
        for (int j = 0; j < 4; ++j) S[j] = z8;
#pragma unroll 4
        for (int c4 = 0; c4 < C_; c4 += 4) {
            const int kcol = c4 + 2 * lh;
            v2f a = *(const v2f*)(Qrow + kcol);
#pragma unroll
            for (int j = 0; j < 4; ++j) {
                v2f bf = *(const v2f*)(&Ktc[(j * 16 + l16) * KT_STRIDE + kcol]);
                S[j] = __builtin_amdgcn_wmma_f32_16x16x4_f32(
                           false, a, false, bf, (short)0, S[j], false, false);
            }
        }

        // ---- raw v_exp_f32 (underflow flush is fine for softmax terms)
#pragma unroll
        for (int j = 0; j < 4; ++j) {
#pragma unroll
            for (int r = 0; r < 8; ++r) {
                float e = __builtin_amdgcn_exp2f(S[j][r] * SCALE_X);
                zacc[r] += e;
                PeP[(wv * 16 + r + 8 * lh) * PT_STRIDE + (j * 16 + l16)] = e;
            }
        }

        // ---- Oe += e @ P_tile ; Ot += ta @ P_tile  (K-dim = BN, chunks of 4)
#pragma unroll 4
        for (int cc = 0; cc < BN; cc += 4) {
            const int kk = cc + 2 * lh;
            v2f ae = *(const v2f*)(&PeP[(wv * 16 + l16) * PT_STRIDE + kk]);
            v2f at = *(const v2f*)(&Tac[(wv * 16 + l16) * PT_STRIDE + kk]);
#pragma unroll
            for (int j = 0; j < 8; ++j) {
                v2f bf;
                bf.x = Ktc[kk * KT_STRIDE + j * 16 + l16];
                bf.y = Ktc[(kk + 1) * KT_STRIDE + j * 16 + l16];
                Oe[j] = __builtin_amdgcn_wmma_f32_16x16x4_f32(
                            false, ae, false, bf, (short)0, Oe[j], false, false);
                Ot[j] = __builtin_amdgcn_wmma_f32_16x16x4_f32(
                            false, at, false, bf, (short)0, Ot[j], false, false);
            }
        }
        __syncthreads();   // Pe/current-buffer reads done before next iteration
    }

    // ---- reduce Z across the 16 lanes of each half (C/D layout: N = lane%16)
#pragma unroll
    for (int r = 0; r < 8; ++r) {
        float z = zacc[r];
        z += __shfl_xor(z, 1, 32);
        z += __shfl_xor(z, 2, 32);
        z += __shfl_xor(z, 4, 32);
        z += __shfl_xor(z, 8, 32);
        zacc[r] = z;
    }

    // ---- packed offset for this batch, then emit valid rows
    long long off = 0;
    for (int i = 0; i < b; ++i) off += lengths[i];
    const long long len_b = lengths[b];

#pragma unroll
    for (int r = 0; r < 8; ++r) {
        const int lr = wv * 16 + r + 8 * lh;      // local row, matches C/D M index
        const int t  = t0 + lr;
        if ((long long)t < len_b) {
            const float rz = 1.0f / zacc[r];
            float* orow = x_out + (size_t)(off + t) * C_;
#pragma unroll
            for (int j = 0; j < 8; ++j)
                orow[j * 16 + l16] = Oe[j][r] * rz + Ot[j][r];
        }
    }
}

// ----------------------------------------------------------------- launch
extern "C" void kernel_launch(void* const* d_in, const int* in_sizes, int n_in,
                              void* d_out, int out_size, void* d_ws, size_t ws_size,
                              hipStream_t stream) {
    const float*     x_temp    = (const float*)d_in[0];
    const float*     x_stat    = (const float*)d_in[1];
    const float*     temp_attn = (const float*)d_in[2];
    // d_in[3] = mask_repeat: provably redundant (padded rows are zero)
    const long long* lengths   = (const long long*)d_in[4];
    const int*       batch_idx = (const int*)d_in[5];
    const int*       pos_idx   = (const int*)d_in[6];
    const float*     W1     = (const float*)d_in[7];
    const float*     b1     = (const float*)d_in[8];
    const float*     gamma1 = (const float*)d_in[9];
    const float*     beta1  = (const float*)d_in[10];
    const float*     mean1  = (const float*)d_in[11];
    const float*     var1   = (const float*)d_in[12];
    const float*     W2     = (const float*)d_in[13];
    const float*     b2     = (const float*)d_in[14];
    const float*     gamma2 = (const float*)d_in[15];
    const float*     beta2  = (const float*)d_in[16];
    const float*     mean2  = (const float*)d_in[17];
    const float*     var2   = (const float*)d_in[18];

    const int N = in_sizes[0] / TEMP_IN;              // 23450 packed rows
    float* padded = (float*)d_ws;                     // [B,T,C] scratch
    float* out    = (float*)d_out;

    const int pad_elems = B_ * T_ * C_;
    zero_kernel<<<(pad_elems + 255) / 256, 256, 0, stream>>>(padded, pad_elems);

    fused_linear_scatter<<<(N + 15) / 16, 128, 0, stream>>>(
        x_temp, x_stat, batch_idx, pos_idx,
        W1, b1, gamma1, beta1, mean1, var1, padded, N);

    stat_kernel<<<B_, STATOUT, 0, stream>>>(
        x_stat, W2, b2, gamma2, beta2, mean2, var2, out + (size_t)N * C_);

    const size_t smem_bytes = (size_t)SMEM_F * sizeof(float);   // ~168 KB
    attn_kernel<<<B_ * 8, 256, smem_bytes, stream>>>(padded, temp_attn, lengths, out);
}